// ST_GAT_52055003627774
// MI455X (gfx1250) — compile-verified
//
#include <hip/hip_runtime.h>
#include <hip/hip_bf16.h>

// ---------------- problem constants (match reference) ----------------
#define NN      65536      // B * N_NODE
#define CC      12         // channels / time steps
#define HH      8          // attention heads
#define EE      1048576    // random edges
#define EPE     (EE + NN)  // edges + self loops = 1114112
#define NNODE   2048
#define BB      32
#define H1      32         // LSTM1 hidden
#define H2      128        // LSTM2 hidden
#define PP      9
#define OUTW    (NNODE * PP)   // 18432

typedef float v2f __attribute__((ext_vector_type(2)));
typedef float v4f __attribute__((ext_vector_type(4)));
typedef float v8f __attribute__((ext_vector_type(8)));

// fp32 WMMA: D(16x16) = A(16x4) * B(4x16) + C   [V_WMMA_F32_16X16X4_F32]
__device__ __forceinline__ v8f wmma4(v2f a, v2f b, v8f c) {
    return __builtin_amdgcn_wmma_f32_16x16x4_f32(
        /*neg_a=*/false, a, /*neg_b=*/false, b,
        /*c_mod=*/(short)0, c, /*reuse_a=*/false, /*reuse_b=*/false);
}

// 8-byte aligned pair load (lowers to global_load_b64 / ds_load_b64)
__device__ __forceinline__ v2f ld2(const float* __restrict__ p) {
    return *(const v2f*)p;
}

__device__ __forceinline__ float sigm(float x) { return 1.0f / (1.0f + __expf(-x)); }

// order-preserving float <-> uint key for atomicMax-based segment max
__device__ __forceinline__ unsigned fkey(float f) {
    unsigned u = __float_as_uint(f);
    return (u & 0x80000000u) ? ~u : (u | 0x80000000u);
}
__device__ __forceinline__ float unfkey(unsigned k) {
    unsigned u = (k & 0x80000000u) ? (k & 0x7fffffffu) : ~k;
    return __uint_as_float(u);
}

// ---------------- init: emax keys, denom, agg ----------------
__global__ void stgat_init_kernel(unsigned* __restrict__ emx,
                                  float* __restrict__ denom,
                                  float* __restrict__ agg) {
    unsigned long long tid = (unsigned long long)blockIdx.x * blockDim.x + threadIdx.x;
    if (tid < (unsigned long long)NN * HH) {
        emx[tid]   = fkey(-1.0e30f);
        denom[tid] = 0.0f;
    }
    if (tid < (unsigned long long)NN * HH * CC) {
        agg[tid] = 0.0f;
    }
}

// ---------------- h = x @ W_gat  ([N,12]@[12,96]) via fp32 WMMA ----------------
__global__ void stgat_gemm_h_kernel(const float* __restrict__ x,
                                    const float* __restrict__ Wg,
                                    float* __restrict__ h) {
    const int wid   = (blockIdx.x * blockDim.x + threadIdx.x) >> 5;  // global wave id
    const int lane  = threadIdx.x & 31;
    const int row16 = lane & 15;
    const int hi    = lane >> 4;
    const int koff  = hi * 2;
    const int tileM = wid / 6;        // 4096 m-tiles
    const int tileN = wid % 6;        // 96 / 16
    const int mrow  = tileM * 16 + row16;
    const int col   = tileN * 16 + row16;

    v8f acc = {};
#pragma unroll
    for (int kb = 0; kb < CC; kb += 4) {
        v2f a = ld2(&x[mrow * CC + kb + koff]);     // even offset -> b64
        v2f b;
        b.x = Wg[(kb + koff) * 96 + col];           // B[k][n] = W_gat row-major [12,96]
        b.y = Wg[(kb + koff + 1) * 96 + col];
        acc = wmma4(a, b, acc);
    }
#pragma unroll
    for (int r = 0; r < 8; ++r)
        h[(tileM * 16 + r + hi * 8) * 96 + col] = acc[r];
}

// ---------------- per-node attention coefficients a_s, a_d ----------------
__global__ void stgat_attn_kernel(const float* __restrict__ h,
                                  const float* __restrict__ att_src,
                                  const float* __restrict__ att_dst,
                                  float* __restrict__ a_s,
                                  float* __restrict__ a_d) {
    int tid = blockIdx.x * blockDim.x + threadIdx.x;
    if (tid >= NN * HH) return;
    int n  = tid >> 3;
    int hh = tid & 7;
    const float* hp = h + n * 96 + hh * CC;
    float s = 0.0f, d = 0.0f;
#pragma unroll
    for (int c = 0; c < CC; ++c) {
        s += hp[c] * att_src[hh * CC + c];
        d += hp[c] * att_dst[hh * CC + c];
    }
    a_s[tid] = s;
    a_d[tid] = d;
}

__device__ __forceinline__ void edge_nodes(const int* __restrict__ ei, int e, int& s, int& d) {
    if (e < EE) { s = ei[e]; d = ei[EE + e]; }
    else        { s = d = e - EE; }   // self loop
}

// ---------------- edge pass 1: segment max ----------------
__global__ void stgat_edge_max_kernel(const int* __restrict__ ei,
                                      const float* __restrict__ a_s,
                                      const float* __restrict__ a_d,
                                      unsigned* __restrict__ emx) {
    unsigned long long tid = (unsigned long long)blockIdx.x * blockDim.x + threadIdx.x;
    if (tid >= (unsigned long long)EPE * HH) return;
    int e  = (int)(tid >> 3);
    int hh = (int)(tid & 7);
    int s, d;
    edge_nodes(ei, e, s, d);
    float v = a_s[s * HH + hh] + a_d[d * HH + hh];
    v = (v > 0.0f) ? v : 0.2f * v;                    // leaky_relu(0.2)
    atomicMax(&emx[d * HH + hh], fkey(v));
}

// ---------------- edge pass 2: segment sum of exp ----------------
__global__ void stgat_edge_sum_kernel(const int* __restrict__ ei,
                                      const float* __restrict__ a_s,
                                      const float* __restrict__ a_d,
                                      const unsigned* __restrict__ emx,
                                      float* __restrict__ denom) {
    unsigned long long tid = (unsigned long long)blockIdx.x * blockDim.x + threadIdx.x;
    if (tid >= (unsigned long long)EPE * HH) return;
    int e  = (int)(tid >> 3);
    int hh = (int)(tid & 7);
    int s, d;
    edge_nodes(ei, e, s, d);
    float v = a_s[s * HH + hh] + a_d[d * HH + hh];
    v = (v > 0.0f) ? v : 0.2f * v;
    float w = __expf(v - unfkey(emx[d * HH + hh]));
    atomicAdd(&denom[d * HH + hh], w);
}

// ---------------- edge pass 3: weighted scatter of h[src] ----------------
// one block of 96 threads per edge; thread idx = h*12 + c
// e = blockIdx.x is wave-uniform, so edge-index loads become scalar loads.
__global__ void stgat_edge_agg_kernel(const int* __restrict__ ei,
                                      const float* __restrict__ a_s,
                                      const float* __restrict__ a_d,
                                      const unsigned* __restrict__ emx,
                                      const float* __restrict__ denom,
                                      const float* __restrict__ h,
                                      float* __restrict__ agg) {
    int e   = blockIdx.x;
    int idx = threadIdx.x;          // 0..95
    int hh  = idx / CC;
    int s, d;
    edge_nodes(ei, e, s, d);
    float v = a_s[s * HH + hh] + a_d[d * HH + hh];
    v = (v > 0.0f) ? v : 0.2f * v;
    float alpha = __expf(v - unfkey(emx[d * HH + hh])) / denom[d * HH + hh];
    atomicAdd(&agg[d * 96 + idx], alpha * h[s * 96 + idx]);
}

// ---------------- head mean + bias -> gT [CC][NN] (transposed for LSTM1) -------
// One thread per node: contiguous 384B agg row in, 12 coalesced stores out.
__global__ void stgat_gat_out_kernel(const float* __restrict__ agg,
                                     const float* __restrict__ bias_gat,
                                     float* __restrict__ gT) {
    int n = blockIdx.x * blockDim.x + threadIdx.x;
    if (n >= NN) return;
    const float* row = agg + (size_t)n * 96;
    float acc[CC];
#pragma unroll
    for (int c = 0; c < CC; ++c) acc[c] = 0.0f;
#pragma unroll
    for (int hh = 0; hh < HH; ++hh) {
#pragma unroll
        for (int c = 0; c < CC; ++c) acc[c] += row[hh * CC + c];
    }
#pragma unroll
    for (int c = 0; c < CC; ++c)
        gT[(size_t)c * NN + n] = acc[c] * (1.0f / HH) + bias_gat[c];
}

// ---------------- LSTM1: persistent single-WG, WMMA gates ----------------
// z[32,128] tiles: 2 m-tiles x 8 n-tiles = 16 waves (512 threads)
// A[b,k] at step t = gT[t*NN + b*2048 + k]  -> contiguous, coalesced b64 loads
__global__ void stgat_lstm1_kernel(const float* __restrict__ gT,
                                   const float* __restrict__ Wih1,
                                   const float* __restrict__ Whh1,
                                   const float* __restrict__ bih1,
                                   const float* __restrict__ bhh1,
                                   float* __restrict__ h1seq) {
    __shared__ float hbuf[BB * H1];     // 1024
    __shared__ float cbuf[BB * H1];
    __shared__ float zbuf[BB * 4 * H1]; // 32x128

    const int tid   = threadIdx.x;
    const int wv    = tid >> 5;
    const int lane  = tid & 31;
    const int row16 = lane & 15;
    const int hi    = lane >> 4;
    const int koff  = hi * 2;

    for (int u = tid; u < BB * H1; u += 512) { hbuf[u] = 0.0f; cbuf[u] = 0.0f; }
    __syncthreads();

    const int mt    = wv >> 3;
    const int nt    = wv & 7;
    const int mbase = mt * 16;
    const int col   = nt * 16 + row16;
    const int arow  = mbase + row16;
    const float bias = bih1[col] + bhh1[col];

    for (int t = 0; t < CC; ++t) {
        v8f acc;
#pragma unroll
        for (int r = 0; r < 8; ++r) acc[r] = bias;

        // x part: K = 2048, contiguous along k in gT
        const float* xrow = gT + (size_t)t * NN + (size_t)arow * NNODE;
        const float* wrow = Wih1 + (size_t)col * NNODE;
        for (int kb = 0; kb < NNODE; kb += 4) {
            v2f a = ld2(&xrow[kb + koff]);
            v2f b = ld2(&wrow[kb + koff]);
            acc = wmma4(a, b, acc);
        }
        // recurrent part: K = 32 from LDS h state
#pragma unroll
        for (int kb = 0; kb < H1; kb += 4) {
            v2f a = ld2(&hbuf[arow * H1 + kb + koff]);
            v2f b = ld2(&Whh1[col * H1 + kb + koff]);
            acc = wmma4(a, b, acc);
        }
#pragma unroll
        for (int r = 0; r < 8; ++r)
            zbuf[(mbase + r + hi * 8) * (4 * H1) + col] = acc[r];
        __syncthreads();

        for (int u = tid; u < BB * H1; u += 512) {
            int b_ = u >> 5, j = u & 31;
            float zi = zbuf[b_ * 128 + j];
            float zf = zbuf[b_ * 128 + 32 + j];
            float zg = zbuf[b_ * 128 + 64 + j];
            float zo = zbuf[b_ * 128 + 96 + j];
            float c  = sigm(zf) * cbuf[u] + sigm(zi) * tanhf(zg);
            float hh = sigm(zo) * tanhf(c);
            cbuf[u] = c;
            hbuf[u] = hh;
            h1seq[t * (BB * H1) + u] = hh;
        }
        __syncthreads();
    }
}

// ---------------- LSTM2: persistent single-WG, dynamic LDS (96KB) ----------------
// z[32,512]: 2 m-tiles x 32 n-tiles = 64 tiles, 16 waves x 4 tiles
__global__ void stgat_lstm2_kernel(const float* __restrict__ h1seq,
                                   const float* __restrict__ Wih2,
                                   const float* __restrict__ Whh2,
                                   const float* __restrict__ bih2,
                                   const float* __restrict__ bhh2,
                                   float* __restrict__ h2last) {
    extern __shared__ float smem[];
    float* zbuf = smem;                        // 32*512 = 16384
    float* hbuf = smem + BB * 4 * H2;          // 32*128 = 4096
    float* cbuf = hbuf + BB * H2;

    const int tid   = threadIdx.x;
    const int wv    = tid >> 5;
    const int lane  = tid & 31;
    const int row16 = lane & 15;
    const int hi    = lane >> 4;
    const int koff  = hi * 2;

    for (int u = tid; u < BB * H2; u += 512) { hbuf[u] = 0.0f; cbuf[u] = 0.0f; }
    __syncthreads();

    for (int t = 0; t < CC; ++t) {
        const float* xs = h1seq + t * (BB * H1);
        for (int tt = wv; tt < 64; tt += 16) {
            int mt = tt >> 5, nt = tt & 31;
            int mbase = mt * 16;
            int col   = nt * 16 + row16;
            int arow  = mbase + row16;
            float bias = bih2[col] + bhh2[col];
            v8f acc;
#pragma unroll
            for (int r = 0; r < 8; ++r) acc[r] = bias;
            // x part: K = 32 from h1seq
#pragma unroll
            for (int kb = 0; kb < H1; kb += 4) {
                v2f a = ld2(&xs[arow * H1 + kb + koff]);
                v2f b = ld2(&Wih2[col * H1 + kb + koff]);
                acc = wmma4(a, b, acc);
            }
            // recurrent: K = 128 from LDS h state
#pragma unroll
            for (int kb = 0; kb < H2; kb += 4) {
                v2f a = ld2(&hbuf[arow * H2 + kb + koff]);
                v2f b = ld2(&Whh2[col * H2 + kb + koff]);
                acc = wmma4(a, b, acc);
            }
#pragma unroll
            for (int r = 0; r < 8; ++r)
                zbuf[(mbase + r + hi * 8) * (4 * H2) + col] = acc[r];
        }
        __syncthreads();

        for (int u = tid; u < BB * H2; u += 512) {
            int b_ = u >> 7, j = u & 127;
            float zi = zbuf[b_ * 512 + j];
            float zf = zbuf[b_ * 512 + 128 + j];
            float zg = zbuf[b_ * 512 + 256 + j];
            float zo = zbuf[b_ * 512 + 384 + j];
            float c  = sigm(zf) * cbuf[u] + sigm(zi) * tanhf(zg);
            float hh = sigm(zo) * tanhf(c);
            cbuf[u] = c;
            hbuf[u] = hh;
            if (t == CC - 1) h2last[u] = hh;
        }
        __syncthreads();
    }
}

// ---------------- final linear: [32,128]@[128,18432] + bias ----------------
__global__ void stgat_final_kernel(const float* __restrict__ h2last,
                                   const float* __restrict__ W_lin,
                                   const float* __restrict__ b_lin,
                                   float* __restrict__ out) {
    const int wid   = (blockIdx.x * blockDim.x + threadIdx.x) >> 5;
    const int lane  = threadIdx.x & 31;
    const int row16 = lane & 15;
    const int hi    = lane >> 4;
    const int koff  = hi * 2;
    const int mt    = wid / (OUTW / 16);   // 0..1
    const int nt    = wid % (OUTW / 16);   // 0..1151
    const int mbase = mt * 16;
    const int col   = nt * 16 + row16;
    const int arow  = mbase + row16;

    float bias = b_lin[col];
    v8f acc;
#pragma unroll
    for (int r = 0; r < 8; ++r) acc[r] = bias;
#pragma unroll
    for (int kb = 0; kb < H2; kb += 4) {
        v2f a = ld2(&h2last[arow * H2 + kb + koff]);
        v2f b = ld2(&W_lin[col * H2 + kb + koff]);
        acc = wmma4(a, b, acc);
    }
#pragma unroll
    for (int r = 0; r < 8; ++r)
        out[(mbase + r + hi * 8) * OUTW + col] = acc[r];
}

// ---------------- host-side orchestration ----------------
extern "C" void kernel_launch(void* const* d_in, const int* in_sizes, int n_in,
                              void* d_out, int out_size, void* d_ws, size_t ws_size,
                              hipStream_t stream) {
    (void)in_sizes; (void)n_in; (void)out_size; (void)ws_size;

    const float* x        = (const float*)d_in[0];
    const int*   ei       = (const int*)  d_in[1];
    const float* W_gat    = (const float*)d_in[2];
    const float* att_src  = (const float*)d_in[3];
    const float* att_dst  = (const float*)d_in[4];
    const float* bias_gat = (const float*)d_in[5];
    const float* Wih1     = (const float*)d_in[6];
    const float* Whh1     = (const float*)d_in[7];
    const float* bih1     = (const float*)d_in[8];
    const float* bhh1     = (const float*)d_in[9];
    const float* Wih2     = (const float*)d_in[10];
    const float* Whh2     = (const float*)d_in[11];
    const float* bih2     = (const float*)d_in[12];
    const float* bhh2     = (const float*)d_in[13];
    const float* W_lin    = (const float*)d_in[14];
    const float* b_lin    = (const float*)d_in[15];
    float* out            = (float*)d_out;

    // workspace layout (floats)
    float*    ws     = (float*)d_ws;
    float*    h      = ws;                                  // 65536*96
    float*    a_s    = h + (size_t)NN * 96;                 // 65536*8
    float*    a_d    = a_s + (size_t)NN * HH;
    unsigned* emx    = (unsigned*)(a_d + (size_t)NN * HH);  // 65536*8
    float*    denom  = (float*)(emx + (size_t)NN * HH);
    float*    agg    = denom + (size_t)NN * HH;             // 65536*96
    float*    gT     = agg + (size_t)NN * 96;               // 12*65536 (transposed)
    float*    h1seq  = gT + (size_t)NN * CC;                // 12*32*32
    float*    h2last = h1seq + (size_t)CC * BB * H1;        // 32*128

    // 1) init emax/denom/agg
    {
        unsigned long long total = (unsigned long long)NN * 96;
        stgat_init_kernel<<<(unsigned)((total + 255) / 256), 256, 0, stream>>>(emx, denom, agg);
    }
    // 2) h = x @ W_gat (WMMA)
    {
        int tiles = (NN / 16) * (96 / 16);  // 24576 wave-tiles, 8 waves/block
        stgat_gemm_h_kernel<<<tiles / 8, 256, 0, stream>>>(x, W_gat, h);
    }
    // 3) attention coefficients
    stgat_attn_kernel<<<(NN * HH + 255) / 256, 256, 0, stream>>>(h, att_src, att_dst, a_s, a_d);
    // 4-6) edge passes
    {
        unsigned long long total = (unsigned long long)EPE * HH;
        unsigned blocks = (unsigned)((total + 255) / 256);
        stgat_edge_max_kernel<<<blocks, 256, 0, stream>>>(ei, a_s, a_d, emx);
        stgat_edge_sum_kernel<<<blocks, 256, 0, stream>>>(ei, a_s, a_d, emx, denom);
        stgat_edge_agg_kernel<<<EPE, 96, 0, stream>>>(ei, a_s, a_d, emx, denom, h, agg);
    }
    // 7) head mean + bias -> transposed gT
    stgat_gat_out_kernel<<<NN / 256, 256, 0, stream>>>(agg, bias_gat, gT);
    // 8) LSTM1 (persistent, 16 waves)
    stgat_lstm1_kernel<<<1, 512, 0, stream>>>(gT, Wih1, Whh1, bih1, bhh1, h1seq);
    // 9) LSTM2 (persistent, dynamic LDS: z 64KB + h/c 32KB)
    stgat_lstm2_kernel<<<1, 512, (BB * 4 * H2 + 2 * BB * H2) * sizeof(float), stream>>>(
        h1seq, Wih2, Whh2, bih2, bhh2, h2last);
    // 10) final linear (WMMA), 2*1152 tiles, 8 waves/block
    {
        int tiles = 2 * (OUTW / 16);  // 2304
        stgat_final_kernel<<<tiles / 8, 256, 0, stream>>>(h2last, W_lin, b_lin, out);
    }
}